// SelectionLayer_5196910428261
// MI455X (gfx1250) — compile-verified
//
#include <hip/hip_runtime.h>

typedef __attribute__((ext_vector_type(2))) float v2f;
typedef __attribute__((ext_vector_type(8))) float v8f;

#define N_EXPERTS 16
#define D_IN      2048
#define D_OUT     2048
#define BATCH     4096
#define KCHUNK    256
#define LDS_STRIDE 260   // 256 + 4 pad: 4-dword bank step per row -> conflict-free
#define MAX_TILES 272    // worst case: 256 tiles (one expert) + 15 singleton tiles
#define NT        4      // 16-wide N-tiles per wave; block covers 8*4*16 = 512 cols

// ---------------- prep kernels ----------------

__global__ void moe_zero_kernel(int* __restrict__ counts, int* __restrict__ fill) {
  int i = threadIdx.x;
  if (i < N_EXPERTS) { counts[i] = 0; fill[i] = 0; }
}

__global__ void moe_count_kernel(const int* __restrict__ actions, int* __restrict__ counts) {
  int i = blockIdx.x * blockDim.x + threadIdx.x;
  if (i < BATCH) atomicAdd(&counts[actions[i] & (N_EXPERTS - 1)], 1);
}

__global__ void moe_tiles_kernel(const int* __restrict__ counts, int* __restrict__ offsets,
                                 int* __restrict__ tileE, int* __restrict__ tileStart,
                                 int* __restrict__ tileValid, int* __restrict__ numTiles) {
  if (threadIdx.x == 0 && blockIdx.x == 0) {
    int off = 0, nt = 0;
    for (int e = 0; e < N_EXPERTS; ++e) {
      offsets[e] = off;
      int c = counts[e];
      for (int t = 0; t < c; t += 16) {
        tileE[nt] = e;
        tileStart[nt] = off + t;
        tileValid[nt] = (c - t < 16) ? (c - t) : 16;
        ++nt;
      }
      off += c;
    }
    numTiles[0] = nt;
  }
}

__global__ void moe_scatter_kernel(const int* __restrict__ actions,
                                   const int* __restrict__ offsets,
                                   int* __restrict__ fill, int* __restrict__ perm) {
  int i = blockIdx.x * blockDim.x + threadIdx.x;
  if (i < BATCH) {
    int e = actions[i] & (N_EXPERTS - 1);
    int pos = offsets[e] + atomicAdd(&fill[e], 1);
    perm[pos] = i;
  }
}

// ---------------- async copy helpers (CDNA5 ASYNCcnt path) ----------------

__device__ __forceinline__ void async_ld_lds_b128(unsigned lds_off, const float* gsrc) {
  // GLOBAL_LOAD_ASYNC_TO_LDS_B128: copies 16B per lane from global address gsrc to LDS offset lds_off,
  // tracked with ASYNCcnt (no VGPR round trip).
  asm volatile("global_load_async_to_lds_b128 %0, %1, off"
               :: "v"(lds_off), "v"(gsrc)
               : "memory");
}

__device__ __forceinline__ void wait_asynccnt0() {
  asm volatile("s_wait_asynccnt 0x0" ::: "memory");
}

// ---------------- grouped GEMM with f32 WMMA + async double-buffered A ----
// block = 256 threads (8 waves). Each block: one 16-token tile of one expert,
// 512 consecutive output columns. Each wave: 4 x (16x16) C tiles.

__global__ __launch_bounds__(256)
void moe_gemm_kernel(const float* __restrict__ xs, const float* __restrict__ W,
                     const float* __restrict__ bias, const int* __restrict__ perm,
                     const int* __restrict__ tileE, const int* __restrict__ tileStart,
                     const int* __restrict__ tileValid, const int* __restrict__ numTiles,
                     float* __restrict__ out) {
  __shared__ float As[2][16][LDS_STRIDE];
  __shared__ int rows[16];

  const int tile = blockIdx.x;
  if (tile >= numTiles[0]) return;        // uniform per block: EXEC stays all-1s for WMMA

  const int e        = tileE[tile];
  const int rowStart = tileStart[tile];
  const int valid    = tileValid[tile];

  const int tid  = threadIdx.x;
  const int lane = tid & 31;
  const int wave = tid >> 5;
  const int lo   = lane & 15;             // N index (B/C/D) and M index (A)
  const int hi   = lane >> 4;             // half-wave select
  const int koff = hi * 2;                // A/B per-lane K offset: {0,2}

  if (tid < 16) rows[tid] = perm[rowStart + (tid < valid ? tid : 0)];
  __syncthreads();

  // Per-thread A staging assignment: row lm, float4 column group c4.
  const int lm = tid >> 4;                // 0..15
  const int c4 = tid & 15;                // 0..15
  const float* srcRow = xs + (size_t)rows[lm] * D_IN;

  const int n0 = blockIdx.y * 512 + wave * 64;

  const float* wbase = W + (size_t)e * D_OUT * D_IN;
  const float* wp[NT];
#pragma unroll
  for (int t = 0; t < NT; ++t)
    wp[t] = wbase + (size_t)(n0 + t * 16 + lo) * D_IN + koff;

  v8f acc[NT];
  const v8f vzero = {0.f, 0.f, 0.f, 0.f, 0.f, 0.f, 0.f, 0.f};
#pragma unroll
  for (int t = 0; t < NT; ++t) acc[t] = vzero;

  // Issue async copies of chunk 0 into buffer 0.
  {
    const unsigned ldsBase = (unsigned)(size_t)&As[0][lm][0];
#pragma unroll
    for (int j = 0; j < 4; ++j) {
      const int cg = c4 + j * 16;         // float4 group 0..63
      async_ld_lds_b128(ldsBase + cg * 16, srcRow + cg * 4);
    }
  }

  int buf = 0;
  for (int k0 = 0; k0 < D_IN; k0 += KCHUNK) {
    wait_asynccnt0();                     // my copies of chunk k0 have landed
    __syncthreads();                      // everyone's copies landed; prev compute done

    if (k0 + KCHUNK < D_IN) {            // prefetch next chunk into the other buffer
      const unsigned ldsBase = (unsigned)(size_t)&As[buf ^ 1][lm][0];
      const float* src = srcRow + (k0 + KCHUNK);
#pragma unroll
      for (int j = 0; j < 4; ++j) {
        const int cg = c4 + j * 16;
        async_ld_lds_b128(ldsBase + cg * 16, src + cg * 4);
      }
    }

    const float* ap = &As[buf][lo][koff];
#pragma unroll 4
    for (int ks = 0; ks < KCHUNK; ks += 4) {
      v2f a = *(const v2f*)(ap + ks);
#pragma unroll
      for (int t = 0; t < NT; ++t) {
        v2f bb = *(const v2f*)(wp[t] + k0 + ks);
        acc[t] = __builtin_amdgcn_wmma_f32_16x16x4_f32(
            /*neg_a=*/false, a, /*neg_b=*/false, bb,
            /*c_mod=*/(short)0, acc[t], /*reuse_a=*/false, /*reuse_b=*/false);
      }
    }
    buf ^= 1;
  }

  // Epilogue: bias add + scatter rows back to original token order.
  const int mhi = hi << 3;                // C/D: VGPR r -> M=r (lanes 0-15) / r+8 (16-31)
#pragma unroll
  for (int t = 0; t < NT; ++t) {
    const int n = n0 + t * 16 + lo;
    const float bv = bias[(size_t)e * D_OUT + n];
#pragma unroll
    for (int r = 0; r < 8; ++r) {
      const int m = mhi + r;
      if (m < valid)
        out[(size_t)rows[m] * D_OUT + n] = acc[t][r] + bv;
    }
  }
}

// ---------------- tuple tail: mxs and actions passthrough ----------------

__global__ void moe_tail_kernel(const float* __restrict__ mxs,
                                const int* __restrict__ actions,
                                float* __restrict__ out) {
  int i = blockIdx.x * blockDim.x + threadIdx.x;
  if (i < BATCH) {
    float* t = out + (size_t)BATCH * D_OUT;
    t[i] = mxs[i];
    t[BATCH + i] = (float)actions[i];
  }
}

// ---------------- launch ----------------

extern "C" void kernel_launch(void* const* d_in, const int* in_sizes, int n_in,
                              void* d_out, int out_size, void* d_ws, size_t ws_size,
                              hipStream_t stream) {
  const float* xs      = (const float*)d_in[0];
  const float* mxs     = (const float*)d_in[1];
  const int*   actions = (const int*)d_in[2];
  const float* W       = (const float*)d_in[3];
  const float* b       = (const float*)d_in[4];
  float* out = (float*)d_out;

  int* counts    = (int*)d_ws;            // 16
  int* fill      = counts + 16;           // 16
  int* offsets   = fill + 16;             // 16
  int* numTiles  = offsets + 16;          // 1 (padded to 16)
  int* perm      = numTiles + 16;         // BATCH
  int* tileE     = perm + BATCH;          // MAX_TILES
  int* tileStart = tileE + MAX_TILES;     // MAX_TILES
  int* tileValid = tileStart + MAX_TILES; // MAX_TILES

  moe_zero_kernel<<<1, 64, 0, stream>>>(counts, fill);
  moe_count_kernel<<<BATCH / 256, 256, 0, stream>>>(actions, counts);
  moe_tiles_kernel<<<1, 1, 0, stream>>>(counts, offsets, tileE, tileStart, tileValid, numTiles);
  moe_scatter_kernel<<<BATCH / 256, 256, 0, stream>>>(actions, offsets, fill, perm);

  dim3 grid(MAX_TILES, D_OUT / 512);      // (272, 4); extra tiles early-exit on numTiles
  moe_gemm_kernel<<<grid, 256, 0, stream>>>(xs, W, b, perm, tileE, tileStart, tileValid,
                                            numTiles, out);

  moe_tail_kernel<<<BATCH / 256, 256, 0, stream>>>(mxs, actions, out);
}